// MultiGraphAttentionLayer_25065429139873
// MI455X (gfx1250) — compile-verified
//
#include <hip/hip_runtime.h>
#include <hip/hip_bf16.h>

// ---------------------------------------------------------------------------
// MultiGraphAttention on MI455X (gfx1250), fp32 throughout.
//
// B=16, N=1024, D=256, HEADS=4, C=64.  Memory-bound: ~150 MB of tensor
// traffic (adj's 67 MB is never read), and with 192 MB L2 the H intermediate
// stays cache-resident, so HBM sees ~x-in + out-out (~34 MB).  GEMMs
// (4.3 GFLOP total) use V_WMMA_F32_16X16X4_F32 for full fp32 accuracy.
// The shared GEMM operand W is staged per-workgroup into LDS with the CDNA5
// async path (GLOBAL_LOAD_ASYNC_TO_LDS_B128 + s_wait_asynccnt), removing the
// 8x-redundant per-wave global loads; B operands then come from LDS
// (ds_load_b64, conflict-free via a 260-float padded row stride).
// ---------------------------------------------------------------------------

typedef float v2f __attribute__((ext_vector_type(2)));
typedef float v8f __attribute__((ext_vector_type(8)));

#define NEG_SLOPE 0.2f
#define WLDS_STRIDE 260   // 256 + 4 pad: 260 % 64 == 4 -> conflict-free banks

// ---------------------------------------------------------------------------
// GEMM: Hout[m,n] = sum_k X[m,k] * W[n,k]   (torch Linear: h = x @ W^T)
// M=16384, N=256, K=256.  Block = 8 waves, all on one 64-wide N-strip;
// wave w handles M-tile (mblock*8 + w).  W strip (64x256) lives in LDS.
// A lane layout (16x4 f32): row M = lane&15, K-pair = 2*(lane>>4)+{0,1}
// B lane layout (4x16 f32): col N = lane&15, same contiguous K-pair.
// ---------------------------------------------------------------------------
__global__ __launch_bounds__(256) void gemm_wmma_kernel(
    const float* __restrict__ X, const float* __restrict__ W,
    float* __restrict__ Hout)
{
    __shared__ float Wlds[64 * WLDS_STRIDE];   // 66,560 B

    const int lane   = threadIdx.x & 31;
    const int wave   = threadIdx.x >> 5;
    const int nstrip = blockIdx.x & 3;          // 4 N-strips of 64
    const int mtile  = (blockIdx.x >> 2) * 8 + wave;  // 1024 M-tiles of 16
    const int nbase  = nstrip << 6;

    // ---- async-stage W[nbase..nbase+63][0..255] into LDS (64 KB, b128/lane)
    for (int it = 0; it < 16; ++it) {
        const int q   = it * 256 + threadIdx.x;     // 0..4095 16B segments
        const int row = q >> 6;
        const int seg = q & 63;
        const float* g = W + (nbase + row) * 256 + seg * 4;
        const unsigned laddr =
            (unsigned)(size_t)(&Wlds[row * WLDS_STRIDE + seg * 4]);
        asm volatile("global_load_async_to_lds_b128 %0, %1, off"
                     :: "v"(laddr), "v"(g) : "memory");
    }
    asm volatile("s_wait_asynccnt 0" ::: "memory");
    __syncthreads();

    const int kh = (lane >> 4) << 1;                // 0 or 2
    const float* xr = X + ((mtile << 4) + (lane & 15)) * 256;
    const float* wl = &Wlds[(lane & 15) * WLDS_STRIDE];

    __builtin_prefetch(xr + 128, 0, 1);             // warm caches for A tail

    v8f acc0 = {}, acc1 = {}, acc2 = {}, acc3 = {};

#pragma unroll 8
    for (int k0 = 0; k0 < 256; k0 += 4) {
        v2f a  = *(const v2f*)(xr + k0 + kh);
        v2f b0 = *(const v2f*)(wl + 0 * 16 * WLDS_STRIDE + k0 + kh);
        v2f b1 = *(const v2f*)(wl + 1 * 16 * WLDS_STRIDE + k0 + kh);
        v2f b2 = *(const v2f*)(wl + 2 * 16 * WLDS_STRIDE + k0 + kh);
        v2f b3 = *(const v2f*)(wl + 3 * 16 * WLDS_STRIDE + k0 + kh);
        acc0 = __builtin_amdgcn_wmma_f32_16x16x4_f32(false, a, false, b0,
                                                     (short)0, acc0, false, false);
        acc1 = __builtin_amdgcn_wmma_f32_16x16x4_f32(false, a, false, b1,
                                                     (short)0, acc1, false, false);
        acc2 = __builtin_amdgcn_wmma_f32_16x16x4_f32(false, a, false, b2,
                                                     (short)0, acc2, false, false);
        acc3 = __builtin_amdgcn_wmma_f32_16x16x4_f32(false, a, false, b3,
                                                     (short)0, acc3, false, false);
    }

    // C/D layout: VGPR r -> row r + 8*(lane>>4), col lane&15
    const int mBase = (mtile << 4) + ((lane >> 4) << 3);
    const int nb    = nbase + (lane & 15);
#pragma unroll
    for (int r = 0; r < 8; ++r) {
        float* o = Hout + (mBase + r) * 256 + nb;
        o[0]  = acc0[r];
        o[16] = acc1[r];
        o[32] = acc2[r];
        o[48] = acc3[r];
    }
}

// ---------------------------------------------------------------------------
// e1/e2 projections: e1[row,k] = sum_d H[row,d] * A[k, (d>>6)*128 + (d&63)]
//                    e2[row,k] = ...                          + 64 ...
// One wave per row; lane owns 8 contiguous features; shfl_xor tree reduce.
// ---------------------------------------------------------------------------
__global__ __launch_bounds__(256) void e_proj_kernel(
    const float* __restrict__ Hm, const float* __restrict__ A,
    float* __restrict__ e1, float* __restrict__ e2)
{
    const int lane = threadIdx.x & 31;
    const int wave = threadIdx.x >> 5;
    const int row  = blockIdx.x * 8 + wave;

    const float* hr = Hm + row * 256 + lane * 8;
    float hv[8];
#pragma unroll
    for (int i = 0; i < 8; ++i) hv[i] = hr[i];

    float s1[4] = {0.f, 0.f, 0.f, 0.f};
    float s2[4] = {0.f, 0.f, 0.f, 0.f};
#pragma unroll
    for (int i = 0; i < 8; ++i) {
        const int d  = lane * 8 + i;
        const int hb = (d >> 6) * 128 + (d & 63);
#pragma unroll
        for (int k = 0; k < 4; ++k) {
            s1[k] = fmaf(hv[i], A[k * 512 + hb], s1[k]);
            s2[k] = fmaf(hv[i], A[k * 512 + hb + 64], s2[k]);
        }
    }
#pragma unroll
    for (int k = 0; k < 4; ++k) {
#pragma unroll
        for (int off = 16; off > 0; off >>= 1) {
            s1[k] += __shfl_xor(s1[k], off, 32);
            s2[k] += __shfl_xor(s2[k], off, 32);
        }
    }
    if (lane < 4) {
        e1[row * 4 + lane] = s1[lane];
        e2[row * 4 + lane] = s2[lane];
    }
}

// ---------------------------------------------------------------------------
// Diagonal softmax weight:
//   logZ[b,i,k] = logsumexp_j lrelu(e1[b,i,k] + e2[b,j,k])   (j over N=1024)
//   d[b,i,k]    = exp(lrelu(e1[b,i,k] + e2[b,i,k]) - logZ[b,i,k])
// Block handles 256 i's of one batch; e2[b,:,:] (16 KB) staged in LDS,
// every thread reads the same LDS float4 per j -> bank broadcast.
// ---------------------------------------------------------------------------
__global__ __launch_bounds__(256) void attn_diag_kernel(
    const float* __restrict__ e1, const float* __restrict__ e2,
    float* __restrict__ dcoef)
{
    __shared__ float e2s[1024 * 4];

    const int b     = blockIdx.x >> 2;
    const int iBase = (blockIdx.x & 3) << 8;

    const float* e2b = e2 + b * 4096;
    for (int t = threadIdx.x; t < 4096; t += 256) e2s[t] = e2b[t];
    __syncthreads();

    const int i  = iBase + threadIdx.x;
    const int gi = b * 1024 + i;

    float ev[4];
#pragma unroll
    for (int k = 0; k < 4; ++k) ev[k] = e1[gi * 4 + k];

    float mx[4] = {-3.0e38f, -3.0e38f, -3.0e38f, -3.0e38f};
    for (int j = 0; j < 1024; ++j) {
        float4 w = ((const float4*)e2s)[j];
        float t0 = ev[0] + w.x, t1 = ev[1] + w.y,
              t2 = ev[2] + w.z, t3 = ev[3] + w.w;
        t0 = t0 >= 0.f ? t0 : NEG_SLOPE * t0;
        t1 = t1 >= 0.f ? t1 : NEG_SLOPE * t1;
        t2 = t2 >= 0.f ? t2 : NEG_SLOPE * t2;
        t3 = t3 >= 0.f ? t3 : NEG_SLOPE * t3;
        mx[0] = fmaxf(mx[0], t0); mx[1] = fmaxf(mx[1], t1);
        mx[2] = fmaxf(mx[2], t2); mx[3] = fmaxf(mx[3], t3);
    }

    float sum[4] = {0.f, 0.f, 0.f, 0.f};
    for (int j = 0; j < 1024; ++j) {
        float4 w = ((const float4*)e2s)[j];
        float t0 = ev[0] + w.x, t1 = ev[1] + w.y,
              t2 = ev[2] + w.z, t3 = ev[3] + w.w;
        t0 = t0 >= 0.f ? t0 : NEG_SLOPE * t0;
        t1 = t1 >= 0.f ? t1 : NEG_SLOPE * t1;
        t2 = t2 >= 0.f ? t2 : NEG_SLOPE * t2;
        t3 = t3 >= 0.f ? t3 : NEG_SLOPE * t3;
        sum[0] += expf(t0 - mx[0]); sum[1] += expf(t1 - mx[1]);
        sum[2] += expf(t2 - mx[2]); sum[3] += expf(t3 - mx[3]);
    }

#pragma unroll
    for (int k = 0; k < 4; ++k) {
        const float logZ = mx[k] + logf(sum[k]);
        float td = ev[k] + e2s[i * 4 + k];
        td = td >= 0.f ? td : NEG_SLOPE * td;
        dcoef[gi * 4 + k] = expf(td - logZ);  // td <= logZ, safe
    }
}

// ---------------------------------------------------------------------------
// out = d * hh   (per-head scale, float4 vectorized; head constant per float4)
// ---------------------------------------------------------------------------
__global__ __launch_bounds__(256) void scale_kernel(
    const float* __restrict__ Hm, const float* __restrict__ dcoef,
    float* __restrict__ out)
{
    const int idx4 = blockIdx.x * 256 + threadIdx.x; // float4 index
    const int row  = idx4 >> 6;                      // 64 float4 per row
    const int h    = (idx4 & 63) >> 4;               // 16 float4 per head
    const float s  = dcoef[row * 4 + h];
    float4 v = ((const float4*)Hm)[idx4];
    float4 r = make_float4(s * v.x, s * v.y, s * v.z, s * v.w);
    ((float4*)out)[idx4] = r;
}

__global__ __launch_bounds__(256) void scale_residual_kernel(
    const float* __restrict__ Hm, const float* __restrict__ dcoef,
    float* __restrict__ out)
{
    const int idx4 = blockIdx.x * 256 + threadIdx.x;
    const int row  = idx4 >> 6;
    const int h    = (idx4 & 63) >> 4;
    const float s  = dcoef[row * 4 + h];
    float4 v = ((const float4*)Hm)[idx4];
    float4 p = ((const float4*)out)[idx4];
    float4 r = make_float4(fmaf(s, v.x, p.x), fmaf(s, v.y, p.y),
                           fmaf(s, v.z, p.z), fmaf(s, v.w, p.w));
    ((float4*)out)[idx4] = r;
}

// ---------------------------------------------------------------------------
extern "C" void kernel_launch(void* const* d_in, const int* in_sizes, int n_in,
                              void* d_out, int out_size, void* d_ws, size_t ws_size,
                              hipStream_t stream) {
    const float* x  = (const float*)d_in[0];
    // d_in[1] = adj : accepted but unused by the reference -> never read
    const float* W0 = (const float*)d_in[2];
    const float* A0 = (const float*)d_in[3];
    const float* W1 = (const float*)d_in[4];
    const float* A1 = (const float*)d_in[5];
    float* out = (float*)d_out;

    float* ws    = (float*)d_ws;
    float* Hbuf  = ws;                  // 16384*256 = 4,194,304 floats (16 MB)
    float* e1    = ws + 4194304;        // 65,536 floats
    float* e2    = e1 + 65536;          // 65,536 floats
    float* dcoef = e2 + 65536;          // 65,536 floats  (total ~17.5 MB)

    const dim3 blk(256);

    // ---- layer 0: h0 = diag_attn(x @ W0^T) -> out
    gemm_wmma_kernel<<<512, blk, 0, stream>>>(x, W0, Hbuf);
    e_proj_kernel  <<<2048, blk, 0, stream>>>(Hbuf, A0, e1, e2);
    attn_diag_kernel<<<64, blk, 0, stream>>>(e1, e2, dcoef);
    scale_kernel   <<<4096, blk, 0, stream>>>(Hbuf, dcoef, out);

    // ---- layer 1: out = diag_attn(out @ W1^T) + out (residual)
    gemm_wmma_kernel<<<512, blk, 0, stream>>>(out, W1, Hbuf);
    e_proj_kernel  <<<2048, blk, 0, stream>>>(Hbuf, A1, e1, e2);
    attn_diag_kernel<<<64, blk, 0, stream>>>(e1, e2, dcoef);
    scale_residual_kernel<<<4096, blk, 0, stream>>>(Hbuf, dcoef, out);
}